// GCN_RNI_64682207478387
// MI455X (gfx1250) — compile-verified
//
#include <hip/hip_runtime.h>

typedef __attribute__((ext_vector_type(2))) float v2f;
typedef __attribute__((ext_vector_type(8))) float v8f;

constexpr int NN   = 50000;   // nodes
constexpr int NNP  = 50016;   // nodes padded to multiple of 32 (GEMM M-tile)
constexpr int NE   = 800000;  // edges
constexpr int DIN  = 128;
constexpr int DRNI = 32;
constexpr int DCAT = 160;     // 128 + 32
constexpr int HD   = 256;     // hidden width (fixed N of every GEMM)

// ---------------------------------------------------------------- utilities
__global__ void k_fill(float* __restrict__ p, float v, int n) {
    int i = blockIdx.x * blockDim.x + threadIdx.x;
    if (i < n) p[i] = v;
}

__global__ void k_deg(const long long* __restrict__ dst, float* __restrict__ deg, int e) {
    int i = blockIdx.x * blockDim.x + threadIdx.x;
    if (i < e) atomicAdd(&deg[(int)dst[i]], 1.0f);
}

__global__ void k_rsqrt(float* __restrict__ p, int n) {
    int i = blockIdx.x * blockDim.x + threadIdx.x;
    if (i < n) p[i] = rsqrtf(p[i]);
}

__global__ void k_concat(const float* __restrict__ x, const float* __restrict__ rni,
                         float* __restrict__ hcat, int total) {
    int i = blockIdx.x * blockDim.x + threadIdx.x;
    if (i < total) {
        int r = i / DCAT, c = i - r * DCAT;
        hcat[i] = (c < DIN) ? x[(size_t)r * DIN + c] : rni[(size_t)r * DRNI + (c - DIN)];
    }
}

// ------------------------------------------------- WMMA fp32 GEMM  C = A*B
// C[M x 256] = A[M x K] * B[K x 256] (+bias)(+relu). A has >= gridDim.x*32 rows,
// K % 4 == 0. Block: 128 threads = 4 waves. Block -> one 32-row strip; wave w
// -> cols [64w, 64w+64). Two M-fragments per wave: the 8 B-loads per k-step
// feed 8 WMMAs (B reuse across row-fragments).
// A-frag (16x4 f32): lane l holds A[l%16][k + (l>>4)*2 .. +1]   (v2f)
// B-frag (4x16 f32): lane l holds B[k+(l>>4)*2][n], B[k+(l>>4)*2+1][n]
// C/D (16x16 f32):   vgpr i, lane l -> row i + ((l>>4)<<3), col l%16
__global__ __launch_bounds__(128)
void k_gemm_wmma(const float* __restrict__ A, const float* __restrict__ B,
                 const float* __restrict__ bias, float* __restrict__ C,
                 int K, int M_store, int use_bias, int use_relu) {
    const int wave  = threadIdx.x >> 5;
    const int lane  = threadIdx.x & 31;
    const int m0    = blockIdx.x * 32;
    const int nbase = wave * 64;
    const int m     = lane & 15;
    const int ncol  = lane & 15;
    const int khalf = (lane >> 4) << 1;   // 0 or 2

    v8f a0f0 = {}, a0f1 = {}, a0f2 = {}, a0f3 = {};
    v8f a1f0 = {}, a1f1 = {}, a1f2 = {}, a1f3 = {};

    const float* Arow0 = A + (size_t)(m0 + m) * K + khalf;
    const float* Arow1 = Arow0 + (size_t)16 * K;
    for (int k = 0; k < K; k += 4) {
        v2f a0 = *(const v2f*)(Arow0 + k);              // 8B-aligned (k even)
        v2f a1 = *(const v2f*)(Arow1 + k);
        const float* B0 = B + (size_t)(k + khalf) * HD;
        const float* B1 = B0 + HD;
        v2f b0, b1, b2, b3;
        b0.x = B0[nbase + ncol];        b0.y = B1[nbase + ncol];
        b1.x = B0[nbase + 16 + ncol];   b1.y = B1[nbase + 16 + ncol];
        b2.x = B0[nbase + 32 + ncol];   b2.y = B1[nbase + 32 + ncol];
        b3.x = B0[nbase + 48 + ncol];   b3.y = B1[nbase + 48 + ncol];
        a0f0 = __builtin_amdgcn_wmma_f32_16x16x4_f32(false, a0, false, b0, (short)0, a0f0, false, false);
        a1f0 = __builtin_amdgcn_wmma_f32_16x16x4_f32(false, a1, false, b0, (short)0, a1f0, false, false);
        a0f1 = __builtin_amdgcn_wmma_f32_16x16x4_f32(false, a0, false, b1, (short)0, a0f1, false, false);
        a1f1 = __builtin_amdgcn_wmma_f32_16x16x4_f32(false, a1, false, b1, (short)0, a1f1, false, false);
        a0f2 = __builtin_amdgcn_wmma_f32_16x16x4_f32(false, a0, false, b2, (short)0, a0f2, false, false);
        a1f2 = __builtin_amdgcn_wmma_f32_16x16x4_f32(false, a1, false, b2, (short)0, a1f2, false, false);
        a0f3 = __builtin_amdgcn_wmma_f32_16x16x4_f32(false, a0, false, b3, (short)0, a0f3, false, false);
        a1f3 = __builtin_amdgcn_wmma_f32_16x16x4_f32(false, a1, false, b3, (short)0, a1f3, false, false);
    }

    const int rofs = (lane >> 4) << 3;    // 0 or 8
    v8f accs[2][4] = {{a0f0, a0f1, a0f2, a0f3}, {a1f0, a1f1, a1f2, a1f3}};
#pragma unroll
    for (int r = 0; r < 2; ++r) {
#pragma unroll
        for (int f = 0; f < 4; ++f) {
            const int col = nbase + f * 16 + ncol;
            const float bv = use_bias ? bias[col] : 0.0f;
#pragma unroll
            for (int i = 0; i < 8; ++i) {
                int row = m0 + r * 16 + rofs + i;
                float v = accs[r][f][i] + bv;
                if (use_relu) v = fmaxf(v, 0.0f);
                if (row < M_store) C[(size_t)row * HD + col] = v;
            }
        }
    }
}

// ----------------------------------------- edge scatter:  agg[dst] += x[src]*norm
// 64 threads per edge, one float4 feature chunk per thread (coalesced gather,
// L2-resident f32 atomics on the scatter side).
__global__ void k_edge_agg(const float4* __restrict__ xf, const float* __restrict__ dis,
                           const long long* __restrict__ src, const long long* __restrict__ dst,
                           float* __restrict__ agg) {
    int idx = blockIdx.x * blockDim.x + threadIdx.x;   // NE*64 = 51.2M < 2^31
    int e = idx >> 6;
    int c = idx & 63;
    if (e < NE) {
        int s = (int)src[e], d = (int)dst[e];
        float nrm = dis[s] * dis[d];
        float4 v = xf[(size_t)s * 64 + c];
        float* out = agg + (size_t)d * HD + c * 4;
        atomicAdd(out + 0, v.x * nrm);
        atomicAdd(out + 1, v.y * nrm);
        atomicAdd(out + 2, v.z * nrm);
        atomicAdd(out + 3, v.w * nrm);
    }
}

// ------------------------------ h = relu(agg + x*dis^2 + b), written over agg
__global__ void k_combine(float* __restrict__ agg, const float* __restrict__ xw,
                          const float* __restrict__ dis, const float* __restrict__ bias,
                          int use_relu) {
    int idx = blockIdx.x * blockDim.x + threadIdx.x;   // NN*64
    if (idx < NN * 64) {
        int node = idx >> 6, c = idx & 63;
        float d2 = dis[node];
        d2 *= d2;
        size_t base = (size_t)node * HD + c * 4;
#pragma unroll
        for (int j = 0; j < 4; ++j) {
            float v = agg[base + j] + xw[base + j] * d2 + bias[c * 4 + j];
            if (use_relu) v = fmaxf(v, 0.0f);
            agg[base + j] = v;
        }
    }
}

// ---------------------------------------------------------------- launcher
extern "C" void kernel_launch(void* const* d_in, const int* in_sizes, int n_in,
                              void* d_out, int out_size, void* d_ws, size_t ws_size,
                              hipStream_t stream) {
    (void)in_sizes; (void)n_in; (void)out_size; (void)ws_size;

    const float*     x    = (const float*)d_in[0];
    const float*     rni  = (const float*)d_in[1];
    const long long* ei   = (const long long*)d_in[2];   // int64 per reference
    const float*     W1   = (const float*)d_in[3];
    const float*     b1   = (const float*)d_in[4];
    const float*     W2   = (const float*)d_in[5];
    const float*     b2   = (const float*)d_in[6];
    const float*     Wo   = (const float*)d_in[7];
    const float*     bo   = (const float*)d_in[8];
    const long long* srcI = ei;          // edge_index[0,:]
    const long long* dstI = ei + NE;     // edge_index[1,:]

    // workspace layout (~135 MB), all big buffers padded to NNP rows
    char*  base = (char*)d_ws;
    float* dis  = (float*)base;                                           // NN f32
    float* hcat = (float*)(base + 256 * 1024);                            // NNP*160
    float* bufA = (float*)(base + 256 * 1024 + (size_t)NNP * DCAT * 4);   // NNP*256
    float* bufB = bufA + (size_t)NNP * HD;                                // NNP*256

    const int TB = 256;
    // degrees with self-loop baked in (init 1.0), then dis = deg^-1/2
    k_fill  <<<(NN + TB - 1) / TB, TB, 0, stream>>>(dis, 1.0f, NN);
    k_deg   <<<(NE + TB - 1) / TB, TB, 0, stream>>>(dstI, dis, NE);
    k_rsqrt <<<(NN + TB - 1) / TB, TB, 0, stream>>>(dis, NN);
    // h = [x | rni]; zero the 16 pad rows so GEMM1 reads no garbage
    k_concat<<<(NN * DCAT + TB - 1) / TB, TB, 0, stream>>>(x, rni, hcat, NN * DCAT);
    k_fill  <<<((NNP - NN) * DCAT + TB - 1) / TB, TB, 0, stream>>>(hcat + (size_t)NN * DCAT, 0.0f, (NNP - NN) * DCAT);

    // ---- layer 1: x1 = hcat @ W1 ; agg ; h1 = relu(agg + x1*dis^2 + b1)
    k_gemm_wmma<<<NNP / 32, 128, 0, stream>>>(hcat, W1, nullptr, bufA, DCAT, NNP, 0, 0);
    k_fill     <<<(NNP * HD) / TB, TB, 0, stream>>>(bufB, 0.0f, NNP * HD);
    k_edge_agg <<<(NE * 64) / TB, TB, 0, stream>>>((const float4*)bufA, dis, srcI, dstI, bufB);
    k_combine  <<<(NN * 64 + TB - 1) / TB, TB, 0, stream>>>(bufB, bufA, dis, b1, 1);

    // ---- layer 2: x2 = h1 @ W2 ; agg ; h2 = relu(agg + x2*dis^2 + b2)
    k_gemm_wmma<<<NNP / 32, 128, 0, stream>>>(bufB, W2, nullptr, bufA, HD, NNP, 0, 0);
    k_fill     <<<(NNP * HD) / TB, TB, 0, stream>>>(bufB, 0.0f, NNP * HD);
    k_edge_agg <<<(NE * 64) / TB, TB, 0, stream>>>((const float4*)bufA, dis, srcI, dstI, bufB);
    k_combine  <<<(NN * 64 + TB - 1) / TB, TB, 0, stream>>>(bufB, bufA, dis, b2, 1);

    // ---- output: out = h2 @ W_out + b_out   (store bounded to NN real rows)
    k_gemm_wmma<<<NNP / 32, 128, 0, stream>>>(bufB, Wo, bo, (float*)d_out, HD, NN, 1, 0);
}